// HungarianMatcher_76922864271401
// MI455X (gfx1250) — compile-verified
//
#include <hip/hip_runtime.h>

typedef float v2f __attribute__((ext_vector_type(2)));
typedef float v8f __attribute__((ext_vector_type(8)));
typedef int   v4i __attribute__((vector_size(16)));   // matches builtin param type

#define LDS_AS __attribute__((address_space(3)))
#define GLB_AS __attribute__((address_space(1)))

// ---------------------------------------------------------------------------
// Kernel 1: per-row softmax over C=92 classes. One wave32 per row.
// probs[row, c] = exp(x - rowmax) / sum  -> written to d_ws (L2-resident).
// ---------------------------------------------------------------------------
__global__ void __launch_bounds__(256)
hm_softmax_rows(const float* __restrict__ logits,
                float* __restrict__ probs, int rows) {
  constexpr int C = 92;
  const int wave = threadIdx.x >> 5;
  const int lane = threadIdx.x & 31;
  const int row  = blockIdx.x * 8 + wave;
  if (row >= rows) return;

  const float* x = logits + (size_t)row * C;
  float v0 = x[lane];
  float v1 = x[lane + 32];
  float v2 = (lane < C - 64) ? x[lane + 64] : -3.402823466e38f;

  float m = fmaxf(fmaxf(v0, v1), v2);
#pragma unroll
  for (int off = 16; off > 0; off >>= 1)
    m = fmaxf(m, __shfl_xor(m, off, 32));

  float e0 = __expf(v0 - m);
  float e1 = __expf(v1 - m);
  float e2 = (lane < C - 64) ? __expf(v2 - m) : 0.0f;
  float s = e0 + e1 + e2;
#pragma unroll
  for (int off = 16; off > 0; off >>= 1)
    s += __shfl_xor(s, off, 32);

  const float inv = 1.0f / s;
  float* p = probs + (size_t)row * C;
  p[lane]      = e0 * inv;
  p[lane + 32] = e1 * inv;
  if (lane < C - 64) p[lane + 64] = e2 * inv;
}

// ---------------------------------------------------------------------------
// Kernel 2: one wave per 16x16 tile of the [BN, T] cost matrix.
//  - async-stage 16 rows x 92 probs into this wave's LDS slice
//  - cost_labels via WMMA f32 16x16x4 against an on-the-fly one-hot B
//    (exact: one nonzero product per K)
//  - L1 + GIoU on VALU in the epilogue, non-temporal streamed output
// ---------------------------------------------------------------------------
__global__ void __launch_bounds__(128)
hm_cost_kernel(const float* __restrict__ probs,
               const float* __restrict__ out_bboxes,
               const int*   __restrict__ tgt_labels,
               const float* __restrict__ tgt_bboxes,
               float* __restrict__ out, int T) {
  constexpr int C = 92;
  constexpr int ROW_F4 = 16 * C / 4;   // 368 float4 per 16-row slice (5888 B)

  const int wave = threadIdx.x >> 5;
  const int lane = threadIdx.x & 31;
  const int r0 = (blockIdx.y * 4 + wave) * 16;   // 225*4 = 900 row tiles
  const int t0 = blockIdx.x * 16;                // 60 col tiles

  __shared__ __align__(16) float sP[4][16 * C];
  float* slice = &sP[wave][0];

  // --- stage probs rows r0..r0+15 (contiguous, 16B aligned) into LDS ------
  const float4* gsrc = (const float4*)(probs + (size_t)r0 * C);
  float4* ldst = (float4*)slice;
  for (int idx = lane; idx < ROW_F4; idx += 32) {
#if __has_builtin(__builtin_amdgcn_global_load_async_to_lds_b128)
    __builtin_amdgcn_global_load_async_to_lds_b128(
        (GLB_AS v4i*)(gsrc + idx),
        (LDS_AS v4i*)(ldst + idx), 0, 0);
#else
    ldst[idx] = gsrc[idx];
#endif
  }
#if __has_builtin(__builtin_amdgcn_s_wait_asynccnt)
  __builtin_amdgcn_s_wait_asynccnt(0);
#else
  asm volatile("s_wait_asynccnt 0" ::: "memory");
#endif
  __syncthreads();

  // --- per-lane tile coordinates (WMMA 16x16 f32 C/D layout) --------------
  const int half = lane >> 4;     // lane half selects M block / K pair
  const int n    = lane & 15;     // column within tile
  const int t    = t0 + n;
  const int tl   = tgt_labels[t];
  const float4 tb = ((const float4*)tgt_bboxes)[t];

  // --- cost_labels gather as P x onehot^T on the matrix pipe --------------
  // A (16x4 f32): VGPR v, half h -> K = v + 2h, M = lane&15
  // B (4x16 f32): VGPR v, half h -> K = v + 2h, N = lane&15
  v8f acc = {0.0f, 0.0f, 0.0f, 0.0f, 0.0f, 0.0f, 0.0f, 0.0f};
#pragma unroll
  for (int k0 = 0; k0 < C; k0 += 4) {
    v2f a, b;
    a.x = slice[n * C + k0 + 2 * half];
    a.y = slice[n * C + k0 + 1 + 2 * half];
    b.x = (tl == (k0 + 2 * half))     ? 1.0f : 0.0f;
    b.y = (tl == (k0 + 1 + 2 * half)) ? 1.0f : 0.0f;
    acc = __builtin_amdgcn_wmma_f32_16x16x4_f32(
        false, a, false, b, (short)0, acc, false, false);
  }

  // --- epilogue: L1 + GIoU per element, stream result ---------------------
  const float area2 = (tb.z - tb.x) * (tb.w - tb.y);
#pragma unroll
  for (int i = 0; i < 8; ++i) {
    const int row = r0 + half * 8 + i;           // D layout: M = 8h + i
    const float4 ob = ((const float4*)out_bboxes)[row];
    const float cx = ob.x, cy = ob.y, w = ob.z, h = ob.w;

    const float l1 = fabsf(cx - tb.x) + fabsf(cy - tb.y) +
                     fabsf(w  - tb.z) + fabsf(h  - tb.w);

    const float x0 = cx - 0.5f * w, y0 = cy - 0.5f * h;
    const float x1 = cx + 0.5f * w, y1 = cy + 0.5f * h;
    const float area1 = w * h;

    const float iw = fmaxf(fminf(x1, tb.z) - fmaxf(x0, tb.x), 0.0f);
    const float ih = fmaxf(fminf(y1, tb.w) - fmaxf(y0, tb.y), 0.0f);
    const float inter = iw * ih;
    const float uni = area1 + area2 - inter;
    const float iou = inter / uni;

    const float ew = fmaxf(fmaxf(x1, tb.z) - fminf(x0, tb.x), 0.0f);
    const float eh = fmaxf(fmaxf(y1, tb.w) - fminf(y0, tb.y), 0.0f);
    const float ae = ew * eh;
    const float giou = iou - (ae - uni) / ae;

    const float cost = (1.0f - acc[i])      // LAMB_LABELS * (1 - prob)
                     + 5.0f * l1            // LAMB_BBOXES * L1
                     + 2.0f * (1.0f - giou);// LAMB_GENIOU * (1 - GIoU)
    __builtin_nontemporal_store(cost, out + (size_t)row * T + t);
  }
}

// ---------------------------------------------------------------------------
extern "C" void kernel_launch(void* const* d_in, const int* in_sizes, int n_in,
                              void* d_out, int out_size, void* d_ws, size_t ws_size,
                              hipStream_t stream) {
  (void)n_in; (void)out_size; (void)ws_size;
  const float* out_labels = (const float*)d_in[0];   // [B,N,C] f32
  const float* out_bboxes = (const float*)d_in[1];   // [B,N,4] f32
  const int*   tgt_labels = (const int*)d_in[2];     // [T] int32
  const float* tgt_bboxes = (const float*)d_in[3];   // [T,4] f32
  float* out = (float*)d_out;                        // [B,N,T] f32

  const int rows = in_sizes[1] / 4;                  // B*N = 14400
  const int T    = in_sizes[2];                      // 960

  float* probs = (float*)d_ws;                       // rows*92 f32 (~5.3 MB)

  // Kernel 1: wave-per-row softmax (8 rows per 256-thread block)
  dim3 b1(256), g1((rows + 7) / 8);
  hm_softmax_rows<<<g1, b1, 0, stream>>>(out_labels, probs, rows);

  // Kernel 2: wave-per-16x16-tile; grid exactly covers 900 x 60 tiles
  dim3 b2(128), g2(T / 16, rows / 64);
  hm_cost_kernel<<<g2, b2, 0, stream>>>(probs, out_bboxes, tgt_labels,
                                        tgt_bboxes, out, T);
}